// FTransGANLocalAttentionBlock_67061619360041
// MI455X (gfx1250) — compile-verified
//
#include <hip/hip_runtime.h>
#include <hip/hip_bf16.h>

// FTransGAN local-attention block for gfx1250 (MI455X), fp32 WMMA path.
// Pipeline: proj(q,k) + proj(v, transposed) -> fused flash-style attention
// (no [N,N] in memory, q tile in LDS, coalesced vT fragments, k prefetch)
// -> pooling MLP -> spatial softmax -> weighted channel sum.

typedef __attribute__((ext_vector_type(2))) float v2f;
typedef __attribute__((ext_vector_type(8))) float v8f;

#define N_PIX 4096
#define C_DIM 512
#define D_DIM 64
#define HID_DIM 100
#define B_DIM 8

__device__ __forceinline__ v8f wmma4f32(v2f a, v2f b, v8f c) {
  // D = A(16x4 f32) * B(4x16 f32) + C(16x16 f32)
  return __builtin_amdgcn_wmma_f32_16x16x4_f32(
      /*neg_a=*/false, a, /*neg_b=*/false, b,
      /*c_mod=*/(short)0, c, /*reuse_a=*/false, /*reuse_b=*/false);
}

// Fragment element (dim16 = l16, K = vgpr + 2*hi), matrix element (m,k) at base[k*stride + m]
__device__ __forceinline__ v2f frag_kstride(const float* base, int stride, int l16, int hi) {
  v2f f;
  f.x = base[(size_t)(2 * hi + 0) * stride + l16];
  f.y = base[(size_t)(2 * hi + 1) * stride + l16];
  return f;
}

// Matrix element (m,k) at base[m*stride + k]
__device__ __forceinline__ v2f frag_mstride(const float* base, int stride, int l16, int hi) {
  const float* p = base + (size_t)l16 * stride + 2 * hi;
  v2f f;
  f.x = p[0];
  f.y = p[1];
  return f;
}

// ---------------------------------------------------------------------------
// Y[b] = W (MxC) * X[b] (CxN) + bias  --- one wave per 16x16 output tile.
// storeT == 0: Y[b] stored [M][N];  storeT == 1: stored transposed [N][M]
// (c-contiguous, so attention A-fragments read coalesced rows).
// ---------------------------------------------------------------------------
__global__ void __launch_bounds__(128) proj_kernel(
    const float* __restrict__ W, const float* __restrict__ bias,
    const float* __restrict__ X, float* __restrict__ Y, int M, int storeT) {
  const int waveId = blockIdx.x * 4 + (threadIdx.x >> 5);
  const int tilesM = M >> 4;
  const int tilesN = N_PIX >> 4;
  const int tilesPerB = tilesM * tilesN;
  const int b = waveId / tilesPerB;
  const int t = waveId % tilesPerB;
  const int m0 = (t / tilesN) << 4;
  const int n0 = (t % tilesN) << 4;

  const float* Xb = X + (size_t)b * C_DIM * N_PIX;
  float* Yb = Y + (size_t)b * M * N_PIX;

  const int lane = threadIdx.x & 31;
  const int l16 = lane & 15;
  const int hi = lane >> 4;

  v8f acc = {};
  for (int c0 = 0; c0 < C_DIM; c0 += 4) {
    v2f a = frag_mstride(W + (size_t)m0 * C_DIM + c0, C_DIM, l16, hi);   // A[m][k]=W[m0+m][c0+k]
    v2f bb = frag_kstride(Xb + (size_t)c0 * N_PIX + n0, N_PIX, l16, hi); // B[k][n]=X[c0+k][n0+n]
    acc = wmma4f32(a, bb, acc);
  }
  // D: M = vgpr + 8*(lane/16), N = lane%16
  if (storeT) {
    float* p = Yb + ((size_t)(n0 + l16) * M + m0 + 8 * hi);  // 8 contiguous c's per lane
#pragma unroll
    for (int v = 0; v < 8; ++v) p[v] = acc[v] + bias[m0 + 8 * hi + v];
  } else {
#pragma unroll
    for (int v = 0; v < 8; ++v) {
      const int m = m0 + v + 8 * hi;
      Yb[(size_t)m * N_PIX + n0 + l16] = acc[v] + bias[m];
    }
  }
}

// ---------------------------------------------------------------------------
// Fused self-attention: one workgroup (8 waves) per (batch, 16-query i-tile).
// Pass 1: online-softmax stats (max, sumexp) over all 4096 keys.
// Pass 2: recompute energy, exp into LDS, accumulate hfeat = gamma*sa/s + x.
// Energy computed transposed (e[j][i] = K^T Q) so j-reduction is lane-local.
// q tile staged once in LDS; v read from transposed layout (coalesced).
// ---------------------------------------------------------------------------
__global__ void __launch_bounds__(256) attn_kernel(
    const float* __restrict__ q, const float* __restrict__ k,
    const float* __restrict__ vt, const float* __restrict__ x,
    const float* __restrict__ gamma, float* __restrict__ hfeat) {
  const int b = blockIdx.x >> 8;           // 256 i-tiles per batch
  const int i0 = (blockIdx.x & 255) << 4;
  const int w = threadIdx.x >> 5;
  const int lane = threadIdx.x & 31;
  const int l16 = lane & 15;
  const int hi = lane >> 4;

  const float* qb = q + (size_t)b * D_DIM * N_PIX;
  const float* kb = k + (size_t)b * D_DIM * N_PIX;
  const float* vtb = vt + (size_t)b * N_PIX * C_DIM;   // vT[j][c]
  const float* xb = x + (size_t)b * C_DIM * N_PIX;
  float* hb = hfeat + (size_t)b * C_DIM * N_PIX;

  __shared__ float qs[D_DIM][16];     // q tile for this i-block (4 KB)
  __shared__ float pbuf[8][16][17];   // 8 waves x (16j x 16i) exp tiles, padded
  __shared__ float redM[8][16];
  __shared__ float redS[8][16];
  __shared__ float rowM[16];
  __shared__ float rowSinv[16];

  // stage q tile: 1024 floats, coalesced
#pragma unroll
  for (int it = 0; it < 4; ++it) {
    const int f = threadIdx.x + (it << 8);
    qs[f >> 4][f & 15] = qb[(size_t)(f >> 4) * N_PIX + i0 + (f & 15)];
  }
  __syncthreads();

  // ---------------- pass 1: softmax statistics ----------------
  float pm = -3.0e38f, ps = 0.0f;     // per-lane running max / sumexp; column i = l16
  for (int jt = w; jt < 256; jt += 8) {
    const int j0 = jt << 4;
    v8f e = {};
#pragma unroll
    for (int d0 = 0; d0 < D_DIM; d0 += 4) {
      v2f a = frag_kstride(kb + (size_t)d0 * N_PIX + j0, N_PIX, l16, hi);  // A[j][d]=k[d][j]
      v2f bb;                                                              // B[d][i]=q[d][i]
      bb.x = qs[d0 + 2 * hi + 0][l16];
      bb.y = qs[d0 + 2 * hi + 1][l16];
      e = wmma4f32(a, bb, e);
    }
    float tmax = pm;
#pragma unroll
    for (int v = 0; v < 8; ++v) tmax = fmaxf(tmax, e[v]);
    float sacc = ps * __expf(pm - tmax);
#pragma unroll
    for (int v = 0; v < 8; ++v) sacc += __expf(e[v] - tmax);
    ps = sacc;
    pm = tmax;
  }
  { // merge lane L with L+16 (they hold disjoint j-halves of the same column i)
    const float om = __shfl_xor(pm, 16, 32);
    const float os = __shfl_xor(ps, 16, 32);
    const float nm = fmaxf(pm, om);
    ps = ps * __expf(pm - nm) + os * __expf(om - nm);
    pm = nm;
  }
  if (lane < 16) { redM[w][l16] = pm; redS[w][l16] = ps; }
  __syncthreads();
  if (w == 0 && lane < 16) {
    float m = redM[0][lane], sacc = redS[0][lane];
#pragma unroll
    for (int ww = 1; ww < 8; ++ww) {
      const float m2 = redM[ww][lane], s2 = redS[ww][lane];
      const float nm = fmaxf(m, m2);
      sacc = sacc * __expf(m - nm) + s2 * __expf(m2 - nm);
      m = nm;
    }
    rowM[lane] = m;
    rowSinv[lane] = 1.0f / sacc;
  }
  __syncthreads();

  // ---------------- pass 2: recompute + accumulate ----------------
  v8f acc[4] = {};                     // 4 c-tiles per wave (c0 = (4w+t)*16)
  const float mi = rowM[l16];
  for (int r = 0; r < 32; ++r) {
    const int j0 = ((r << 3) + w) << 4;
    if (r < 31) {
      // prefetch next round's k tile: one row per lane (16 floats/row = 1 line)
      const int jn = (((r + 1) << 3) + w) << 4;
      __builtin_prefetch(kb + (size_t)lane * N_PIX + jn, 0, 1);
      __builtin_prefetch(kb + (size_t)(lane + 32) * N_PIX + jn, 0, 1);
    }
    v8f e = {};
#pragma unroll
    for (int d0 = 0; d0 < D_DIM; d0 += 4) {
      v2f a = frag_kstride(kb + (size_t)d0 * N_PIX + j0, N_PIX, l16, hi);
      v2f bb;
      bb.x = qs[d0 + 2 * hi + 0][l16];
      bb.y = qs[d0 + 2 * hi + 1][l16];
      e = wmma4f32(a, bb, e);
    }
#pragma unroll
    for (int v = 0; v < 8; ++v)
      pbuf[w][v + 8 * hi][l16] = __expf(e[v] - mi);   // e element (j=v+8*hi, i=l16)
    __syncthreads();
#pragma unroll
    for (int t = 0; t < 4; ++t) {
      const int c0 = ((w << 2) + t) << 4;
#pragma unroll
      for (int sub = 0; sub < 8; ++sub) {
        const int js = ((r << 3) + sub) << 4;
#pragma unroll
        for (int kc = 0; kc < 4; ++kc) {
          // A[c][jk] = vT[js + kc*4 + jk][c0+c]  (coalesced: lanes 0..15 contiguous c)
          const float* vp = vtb + (size_t)(js + (kc << 2) + 2 * hi) * C_DIM + c0 + l16;
          v2f a;
          a.x = vp[0];
          a.y = vp[C_DIM];
          // B[jk][i] = p[js + kc*4 + jk][i]
          v2f bb;
          bb.x = pbuf[sub][(kc << 2) + 2 * hi + 0][l16];
          bb.y = pbuf[sub][(kc << 2) + 2 * hi + 1][l16];
          acc[t] = wmma4f32(a, bb, acc[t]);
        }
      }
    }
    __syncthreads();
  }

  const float g = gamma[0];
  const float si = rowSinv[l16];
#pragma unroll
  for (int t = 0; t < 4; ++t) {
    const int c0 = ((w << 2) + t) << 4;
#pragma unroll
    for (int v = 0; v < 8; ++v) {
      const int c = c0 + v + 8 * hi;
      const size_t idx = (size_t)c * N_PIX + i0 + l16;
      hb[idx] = g * (acc[t][v] * si) + xb[idx];
    }
  }
}

// ---------------------------------------------------------------------------
// score[b,n] = Wc . tanh(Wa @ hfeat[b,:,n] + ba) --- one thread per pixel.
// ---------------------------------------------------------------------------
__global__ void __launch_bounds__(256) pool_score_kernel(
    const float* __restrict__ Wa, const float* __restrict__ ba,
    const float* __restrict__ Wc, const float* __restrict__ hfeat,
    float* __restrict__ score) {
  const int idx = blockIdx.x * 256 + threadIdx.x;
  const int b = idx >> 12;
  const int n = idx & (N_PIX - 1);
  const float* hb = hfeat + (size_t)b * C_DIM * N_PIX + n;
  float a[HID_DIM];
#pragma unroll
  for (int h = 0; h < HID_DIM; ++h) a[h] = ba[h];
  for (int c = 0; c < C_DIM; ++c) {
    const float xv = hb[(size_t)c * N_PIX];
#pragma unroll
    for (int h = 0; h < HID_DIM; ++h)
      a[h] = fmaf(Wa[h * C_DIM + c], xv, a[h]);   // lane-uniform weight -> scalar loads
  }
  float s = 0.f;
#pragma unroll
  for (int h = 0; h < HID_DIM; ++h) s += Wc[h] * tanhf(a[h]);
  score[idx] = s;
}

// ---------------------------------------------------------------------------
// amap[b,:] = softmax(score[b,:]) over N --- one block per batch.
// ---------------------------------------------------------------------------
__global__ void __launch_bounds__(256) softmax_kernel(
    const float* __restrict__ score, float* __restrict__ amap) {
  const int b = blockIdx.x;
  const float* s = score + (size_t)b * N_PIX;
  __shared__ float wred[8];
  const int lane = threadIdx.x & 31;
  const int w = threadIdx.x >> 5;

  float m = -3.0e38f;
  for (int n = threadIdx.x; n < N_PIX; n += 256) m = fmaxf(m, s[n]);
#pragma unroll
  for (int off = 16; off > 0; off >>= 1) m = fmaxf(m, __shfl_xor(m, off, 32));
  if (lane == 0) wred[w] = m;
  __syncthreads();
  m = wred[0];
#pragma unroll
  for (int i = 1; i < 8; ++i) m = fmaxf(m, wred[i]);
  __syncthreads();

  float sum = 0.f;
  for (int n = threadIdx.x; n < N_PIX; n += 256) sum += __expf(s[n] - m);
#pragma unroll
  for (int off = 16; off > 0; off >>= 1) sum += __shfl_xor(sum, off, 32);
  if (lane == 0) wred[w] = sum;
  __syncthreads();
  float tot = 0.f;
#pragma unroll
  for (int i = 0; i < 8; ++i) tot += wred[i];
  const float inv = 1.0f / tot;
  for (int n = threadIdx.x; n < N_PIX; n += 256)
    amap[(size_t)b * N_PIX + n] = __expf(s[n] - m) * inv;
}

// ---------------------------------------------------------------------------
// out[b,c] = sum_n x[b,c,n] * amap[b,n] --- one wave per (b,c).
// ---------------------------------------------------------------------------
__global__ void __launch_bounds__(256) out_kernel(
    const float* __restrict__ x, const float* __restrict__ amap,
    float* __restrict__ out) {
  const int wid = (blockIdx.x * blockDim.x + threadIdx.x) >> 5;
  const int lane = threadIdx.x & 31;
  const int b = wid >> 9;
  const int c = wid & (C_DIM - 1);
  const float* xr = x + ((size_t)b * C_DIM + c) * N_PIX;
  const float* am = amap + (size_t)b * N_PIX;
  float s = 0.f;
  for (int n = lane; n < N_PIX; n += 32) s = fmaf(xr[n], am[n], s);
#pragma unroll
  for (int off = 16; off > 0; off >>= 1) s += __shfl_xor(s, off, 32);
  if (lane == 0) out[(size_t)b * C_DIM + c] = s;
}

// ---------------------------------------------------------------------------
extern "C" void kernel_launch(void* const* d_in, const int* in_sizes, int n_in,
                              void* d_out, int out_size, void* d_ws, size_t ws_size,
                              hipStream_t stream) {
  (void)in_sizes; (void)n_in; (void)out_size; (void)ws_size;

  const float* x     = (const float*)d_in[0];
  const float* Wq    = (const float*)d_in[1];
  const float* bq    = (const float*)d_in[2];
  const float* Wk    = (const float*)d_in[3];
  const float* bk    = (const float*)d_in[4];
  const float* Wv    = (const float*)d_in[5];
  const float* bv    = (const float*)d_in[6];
  const float* gamma = (const float*)d_in[7];
  const float* Wa    = (const float*)d_in[8];
  const float* ba    = (const float*)d_in[9];
  const float* Wc    = (const float*)d_in[10];
  float* out = (float*)d_out;

  float* ws = (float*)d_ws;
  float* qbuf  = ws;                                          // 8*64*4096
  float* kbuf  = qbuf  + (size_t)B_DIM * D_DIM * N_PIX;       // 8*64*4096
  float* vtbuf = kbuf  + (size_t)B_DIM * D_DIM * N_PIX;       // 8*4096*512 (transposed)
  float* hfeat = vtbuf + (size_t)B_DIM * C_DIM * N_PIX;       // 8*512*4096
  float* score = hfeat + (size_t)B_DIM * C_DIM * N_PIX;       // 8*4096
  float* amap  = score + (size_t)B_DIM * N_PIX;               // 8*4096

  // q, k projections: 8 * 4 * 256 = 8192 waves, 4 waves per block
  {
    const int waves = B_DIM * (D_DIM / 16) * (N_PIX / 16);
    proj_kernel<<<waves / 4, 128, 0, stream>>>(Wq, bq, x, qbuf, D_DIM, 0);
    proj_kernel<<<waves / 4, 128, 0, stream>>>(Wk, bk, x, kbuf, D_DIM, 0);
  }
  // v projection (stored transposed): 8 * 32 * 256 = 65536 waves
  {
    const int waves = B_DIM * (C_DIM / 16) * (N_PIX / 16);
    proj_kernel<<<waves / 4, 128, 0, stream>>>(Wv, bv, x, vtbuf, C_DIM, 1);
  }
  // fused attention: one block per (batch, 16-query tile)
  attn_kernel<<<B_DIM * (N_PIX / 16), 256, 0, stream>>>(qbuf, kbuf, vtbuf, x, gamma, hfeat);
  // pooling MLP score
  pool_score_kernel<<<(B_DIM * N_PIX) / 256, 256, 0, stream>>>(Wa, ba, Wc, hfeat, score);
  // spatial softmax
  softmax_kernel<<<B_DIM, 256, 0, stream>>>(score, amap);
  // weighted sum of original features
  out_kernel<<<(B_DIM * C_DIM) / 8, 256, 0, stream>>>(x, amap, out);
}